// JointTrainingModule_40261023433020
// MI455X (gfx1250) — compile-verified
//
#include <hip/hip_runtime.h>
#include <math.h>

typedef __attribute__((ext_vector_type(2))) float v2f;
typedef __attribute__((ext_vector_type(4))) float v4f;
typedef __attribute__((ext_vector_type(8))) float v8f;

#define BQ   64
#define NG   2048
#define DIM  128
#define TAU  0.1f
#define EPSN 1e-8f

// ws layout (floats): rnq[64] | rng[2048] | idx0[64](int) idx1[64](int) | w0[64] w1[64]
#define OFF_RNQ 0
#define OFF_RNG 64
#define OFF_TOP (64 + 2048)

// gather geometry (in float4 units)
#define IMG4 49152   // 3*256*256/4 per gallery image
#define MSK4 16384   // 256*256/4 per gallery mask
#define OUT_MASK4 ((size_t)BQ * IMG4)  // float4 offset of mask region in d_out

// -------- 1) inverse norms for q rows and g rows ----------------------------
__global__ void __launch_bounds__(128) norms_kernel(const float* __restrict__ q,
                                                    const float* __restrict__ g,
                                                    float* __restrict__ ws) {
    int wave = (blockIdx.x * blockDim.x + threadIdx.x) >> 5;
    int lane = threadIdx.x & 31;
    if (wave >= BQ + NG) return;
    const float* src = (wave < BQ) ? (q + (size_t)wave * DIM)
                                   : (g + (size_t)(wave - BQ) * DIM);
    v4f v = *(const v4f*)(src + lane * 4);                // 128 floats / 32 lanes
    float ss = v.x * v.x + v.y * v.y + v.z * v.z + v.w * v.w;
    #pragma unroll
    for (int off = 16; off >= 1; off >>= 1) ss += __shfl_xor(ss, off, 32);
    if (lane == 0) {
        float rn = 1.0f / fmaxf(sqrtf(ss), EPSN);
        if (wave < BQ) ws[OFF_RNQ + wave] = rn;
        else           ws[OFF_RNG + (wave - BQ)] = rn;
    }
}

// -------- 2) raw dot products via V_WMMA_F32_16X16X4_F32 --------------------
// One wave per 16x16 output tile; sim (B x N, raw dots) stored in d_out scratch.
__global__ void __launch_bounds__(32) sim_wmma_kernel(const float* __restrict__ q,
                                                      const float* __restrict__ g,
                                                      float* __restrict__ sim) {
    const int n0   = blockIdx.x * 16;
    const int m0   = blockIdx.y * 16;
    const int lane = threadIdx.x;
    const int row  = lane & 15;
    const int kk   = (lane >> 4) * 2;   // lanes 0-15 -> K pair {0,1}; 16-31 -> {2,3}

    const float* qa = q + (size_t)(m0 + row) * DIM + kk;
    const float* gb = g + (size_t)(n0 + row) * DIM + kk;   // B[k][n]=g[n][k]: same pattern

    v8f c = {};
    #pragma unroll 4
    for (int k0 = 0; k0 < DIM; k0 += 4) {
        v2f a = *(const v2f*)(qa + k0);
        v2f b = *(const v2f*)(gb + k0);
        c = __builtin_amdgcn_wmma_f32_16x16x4_f32(false, a, false, b,
                                                  (short)0, c, false, false);
    }
    // C/D layout: VGPR r -> row m0+r (lanes 0-15) / m0+r+8 (lanes 16-31)
    const int col = n0 + (lane & 15);
    const int mb  = m0 + (lane >> 4) * 8;
    #pragma unroll
    for (int r = 0; r < 8; ++r) sim[(size_t)(mb + r) * NG + col] = c[r];
}

// -------- 3) top-2 + temperature softmax per batch row ----------------------
__global__ void __launch_bounds__(32) top2_kernel(const float* __restrict__ sim,
                                                  float* __restrict__ ws) {
    const int b    = blockIdx.x;
    const int lane = threadIdx.x;
    const float* row = sim + (size_t)b * NG;
    const float* rng = ws + OFF_RNG;
    const float  rq  = ws[OFF_RNQ + b];

    float v0 = -3.4e38f, v1 = -3.4e38f;
    int   i0 = 0x7fffffff, i1 = 0x7fffffff;
    for (int n = lane; n < NG; n += 32) {
        float s = row[n] * rq * rng[n];               // cosine similarity
        if (s > v0 || (s == v0 && n < i0)) { v1 = v0; i1 = i0; v0 = s; i0 = n; }
        else if (s > v1 || (s == v1 && n < i1)) { v1 = s; i1 = n; }
    }
    #pragma unroll
    for (int off = 16; off >= 1; off >>= 1) {
        float u0 = __shfl_xor(v0, off, 32); int j0 = __shfl_xor(i0, off, 32);
        float u1 = __shfl_xor(v1, off, 32); int j1 = __shfl_xor(i1, off, 32);
        if (u0 > v0 || (u0 == v0 && j0 < i0)) {
            float nv1; int ni1;
            if (v0 > u1 || (v0 == u1 && i0 < j1)) { nv1 = v0; ni1 = i0; }
            else                                  { nv1 = u1; ni1 = j1; }
            v0 = u0; i0 = j0; v1 = nv1; i1 = ni1;
        } else if (u0 > v1 || (u0 == v1 && j0 < i1)) {
            v1 = u0; i1 = j0;
        }
    }
    if (lane == 0) {
        float e  = __expf((v1 - v0) / TAU);           // <= 1, numerically stable
        float w0 = 1.0f / (1.0f + e);
        int*   ti = (int*)(ws + OFF_TOP);
        float* tw = ws + OFF_TOP + 2 * BQ;
        ti[b]      = i0;  ti[BQ + b] = i1;
        tw[b]      = w0;  tw[BQ + b] = 1.0f - w0;
    }
}

// -------- 4) bandwidth-bound weighted gather (the ~8 us roofline phase) -----
__global__ void __launch_bounds__(256) gather_kernel(const float* __restrict__ img,
                                                     const float* __restrict__ msk,
                                                     const float* __restrict__ ws,
                                                     float* __restrict__ out) {
    const int b = blockIdx.y;
    const int* ti = (const int*)(ws + OFF_TOP);
    const float* tw = ws + OFF_TOP + 2 * BQ;
    const int   i0 = ti[b], i1 = ti[BQ + b];          // uniform per block -> SGPRs
    const float w0 = tw[b], w1 = tw[BQ + b];

    const int f = blockIdx.x * 256 + threadIdx.x;     // float4 index, [0, IMG4+MSK4)
    const v4f* img4 = (const v4f*)img;
    const v4f* msk4 = (const v4f*)msk;
    v4f*       out4 = (v4f*)out;

    if (f < IMG4) {                                   // uniform branch (192 vs 64 blocks)
        v4f a = img4[(size_t)i0 * IMG4 + f];
        v4f c = img4[(size_t)i1 * IMG4 + f];
        v4f r = w0 * a + w1 * c;
        __builtin_nontemporal_store(r, &out4[(size_t)b * IMG4 + f]);
    } else {
        const int fm = f - IMG4;
        v4f a = msk4[(size_t)i0 * MSK4 + fm];
        v4f c = msk4[(size_t)i1 * MSK4 + fm];
        v4f r = w0 * a + w1 * c;
        __builtin_nontemporal_store(r, &out4[OUT_MASK4 + (size_t)b * MSK4 + fm]);
    }
}

extern "C" void kernel_launch(void* const* d_in, const int* in_sizes, int n_in,
                              void* d_out, int out_size, void* d_ws, size_t ws_size,
                              hipStream_t stream) {
    const float* q   = (const float*)d_in[0];   // (64,128)
    const float* g   = (const float*)d_in[1];   // (2048,128)
    const float* img = (const float*)d_in[2];   // (2048,3,256,256)
    const float* msk = (const float*)d_in[3];   // (2048,256,256)
    float* out = (float*)d_out;                 // image(64,3,256,256) ++ mask(64,256,256)
    float* ws  = (float*)d_ws;                  // ~10 KB of small per-row state
    float* sim = out;                           // 64x2048 raw dots: dead scratch in d_out,
                                                // fully overwritten by gather_kernel later

    norms_kernel<<<(BQ + NG) / 4, 128, 0, stream>>>(q, g, ws);
    sim_wmma_kernel<<<dim3(NG / 16, BQ / 16), 32, 0, stream>>>(q, g, sim);
    top2_kernel<<<BQ, 32, 0, stream>>>(sim, ws);
    gather_kernel<<<dim3((IMG4 + MSK4) / 256, BQ), 256, 0, stream>>>(img, msk, ws, out);
}